// LSTMNet_31464930411041
// MI455X (gfx1250) — compile-verified
//
#include <hip/hip_runtime.h>
#include <math.h>

typedef float v2f __attribute__((ext_vector_type(2)));
typedef float v8f __attribute__((ext_vector_type(8)));

#define BN_EPS 1e-5f

// ---------------------------------------------------------------------------
// WMMA helper: D = A(16x4 f32) * B(4x16 f32) + C(16x16 f32), wave32.
// A frag: lane L holds A[L%16][2*(L/16) + v], v in {0,1}
// B frag: lane L holds B[v + 2*(L/16)][L%16]
// C/D:    lane L, vgpr v holds D[v + 8*(L/16)][L%16]
// ---------------------------------------------------------------------------
static __device__ __forceinline__ v8f wmma4(v2f a, v2f b, v8f c) {
  return __builtin_amdgcn_wmma_f32_16x16x4_f32(false, a, false, b, (short)0, c,
                                               false, false);
}

static __device__ __forceinline__ float sigf(float v) {
  return 1.f / (1.f + __expf(-v));
}
static __device__ __forceinline__ float tanhfast(float v) {
  float e = __expf(-2.f * fabsf(v));
  float t = (1.f - e) / (1.f + e);
  return copysignf(t, v);
}

// Build W1^T fragments (K padded to 12, bias folded at k==9).
static __device__ __forceinline__ void buildB1(const float* __restrict__ W1,
                                               const float* __restrict__ b1,
                                               int lm, int half, v2f bf[3][2]) {
#pragma unroll
  for (int kc = 0; kc < 3; kc++) {
#pragma unroll
    for (int nb = 0; nb < 2; nb++) {
      int n = nb * 16 + lm;
      int k0 = kc * 4 + half * 2;
      int k1 = k0 + 1;
      float vx = 0.f, vy = 0.f;
      if (n < 18) {
        vx = (k0 < 9) ? W1[n * 9 + k0] : 0.f;
        vy = (k1 < 9) ? W1[n * 9 + k1] : ((k1 == 9) ? b1[n] : 0.f);
      }
      bf[kc][nb].x = vx;
      bf[kc][nb].y = vy;
    }
  }
}

// Build W2^T fragments (K padded to 20, bias folded at k==18).
static __device__ __forceinline__ void buildB2(const float* __restrict__ W2,
                                               const float* __restrict__ b2,
                                               int lm, int half, v2f bf[5]) {
#pragma unroll
  for (int kc = 0; kc < 5; kc++) {
    int n = lm;
    int k0 = kc * 4 + half * 2;
    int k1 = k0 + 1;
    float vx = 0.f, vy = 0.f;
    if (n < 9) {
      vx = (k0 < 18) ? W2[n * 18 + k0] : ((k0 == 18) ? b2[n] : 0.f);
      vy = (k1 < 18) ? W2[n * 18 + k1] : 0.f;
    }
    bf[kc].x = vx;
    bf[kc].y = vy;
  }
}

// GEMM1 for one wave: 32 samples (2 M-tiles of 16), z1 = h @ W1^T + b1.
// acc[t][nb] holds 16x16 D tile (nb=0: channels 0..15, nb=1: 16..17).
// Invalid samples (s>=B) produce exactly z=0 (bias is folded into A).
static __device__ __forceinline__ void gemm1(const float* __restrict__ x,
                                             const float* __restrict__ emb,
                                             long long B, long long sbase,
                                             int lm, int half,
                                             const v2f bf[3][2],
                                             v8f acc[2][2]) {
#pragma unroll
  for (int t = 0; t < 2; t++) {
    long long s = sbase + t * 16 + lm;
    float m = (s < B) ? 1.f : 0.f;
    long long sc = (s < B) ? s : (B - 1);
    const float4* xr = (const float4*)(x + sc * 8);
    float4 q0 = xr[0];
    float4 q1 = xr[1];
    int idx = ((int)q0.x) & 7;
    float hv[12];
    hv[0] = emb[idx * 2 + 0] * m;
    hv[1] = emb[idx * 2 + 1] * m;
    hv[2] = q0.y * m;
    hv[3] = q0.z * m;
    hv[4] = q0.w * m;
    hv[5] = q1.x * m;
    hv[6] = q1.y * m;
    hv[7] = q1.z * m;
    hv[8] = q1.w * m;
    hv[9] = m;   // bias column
    hv[10] = 0.f;
    hv[11] = 0.f;
#pragma unroll
    for (int kc = 0; kc < 3; kc++) {
      v2f a;
      a.x = half ? hv[kc * 4 + 2] : hv[kc * 4 + 0];
      a.y = half ? hv[kc * 4 + 3] : hv[kc * 4 + 1];
#pragma unroll
      for (int nb = 0; nb < 2; nb++) acc[t][nb] = wmma4(a, bf[kc][nb], acc[t][nb]);
    }
  }
}

// ---------------------------------------------------------------------------
// Pass 1: z1 stats (sum, sumsq per 18 channels) -> deterministic partials
// ---------------------------------------------------------------------------
__global__ void __launch_bounds__(256) k_pass1(const float* __restrict__ x,
                                               const float* __restrict__ emb,
                                               const float* __restrict__ W1,
                                               const float* __restrict__ b1,
                                               float* __restrict__ P1,
                                               long long B, int tiles) {
  __shared__ float red[8 * 36];
  int tid = threadIdx.x, lane = tid & 31, wave = tid >> 5;
  int lm = lane & 15, half = lane >> 4;
  v2f bf[3][2];
  buildB1(W1, b1, lm, half, bf);
  float psum[2] = {0.f, 0.f}, psq[2] = {0.f, 0.f};
  for (int tile = blockIdx.x; tile < tiles; tile += gridDim.x) {
    long long sbase = (long long)tile * 256 + wave * 32;
    v8f acc[2][2] = {{{0,0,0,0,0,0,0,0},{0,0,0,0,0,0,0,0}},
                     {{0,0,0,0,0,0,0,0},{0,0,0,0,0,0,0,0}}};
    gemm1(x, emb, B, sbase, lm, half, bf, acc);
#pragma unroll
    for (int t = 0; t < 2; t++)
#pragma unroll
      for (int nb = 0; nb < 2; nb++)
#pragma unroll
        for (int v = 0; v < 8; v++) {
          float z = acc[t][nb][v];
          psum[nb] += z;
          psq[nb] += z * z;
        }
  }
#pragma unroll
  for (int nb = 0; nb < 2; nb++) {
    float s = psum[nb] + __shfl_xor(psum[nb], 16, 32);
    float q = psq[nb] + __shfl_xor(psq[nb], 16, 32);
    int n = nb * 16 + lm;
    if (half == 0 && n < 18) {
      red[wave * 36 + n] = s;
      red[wave * 36 + 18 + n] = q;
    }
  }
  __syncthreads();
  if (tid < 36) {
    float tot = 0.f;
    for (int w = 0; w < 8; w++) tot += red[w * 36 + tid];
    P1[(long long)blockIdx.x * 36 + tid] = tot;
  }
}

// ---------------------------------------------------------------------------
// Finalize: reduce partials (fixed order) -> scale/shift (generic over C)
// P stride = 2C: sums at [0,C), sumsq at [C,2C)
// ---------------------------------------------------------------------------
__global__ void k_fin(const float* __restrict__ P, const float* __restrict__ g,
                      const float* __restrict__ be, float* __restrict__ S,
                      int nblk, long long B, int C) {
  int c = threadIdx.x;
  if (c < C) {
    float s = 0.f, q = 0.f;
    for (int b = 0; b < nblk; b++) {
      s += P[(long long)b * (2 * C) + c];
      q += P[(long long)b * (2 * C) + C + c];
    }
    float inv = 1.f / (float)B;
    float mu = s * inv;
    float var = q * inv - mu * mu;
    float sc = g[c] * rsqrtf(var + BN_EPS);
    S[c] = sc;
    S[C + c] = be[c] - mu * sc;
  }
}

// Shared piece: apply BN1+relu to acc1 and stage h1 rows into LDS (stride 21,
// bias column at slot 18, zero pad at 19/20). Invalid rows are zeroed.
static __device__ __forceinline__ void stageH1(float* __restrict__ stage,
                                               const v8f acc[2][2],
                                               float sc1a, float sh1a,
                                               float sc1b, float sh1b,
                                               long long sbase0, long long B,
                                               int wave, int lm, int half) {
#pragma unroll
  for (int t = 0; t < 2; t++) {
#pragma unroll
    for (int v = 0; v < 8; v++) {
      int r = wave * 32 + t * 16 + v + half * 8;
      long long s = sbase0 + r;
      float m = (s < B) ? 1.f : 0.f;
      {
        float val = acc[t][0][v] * sc1a + sh1a;
        stage[r * 21 + lm] = fmaxf(val, 0.f) * m;
      }
      int n = 16 + lm;
      if (n < 18) {
        float val = acc[t][1][v] * sc1b + sh1b;
        stage[r * 21 + n] = fmaxf(val, 0.f) * m;
      } else if (n == 18) {
        stage[r * 21 + 18] = m;  // bias column for GEMM2
      } else if (n <= 20) {
        stage[r * 21 + n] = 0.f;
      }
    }
  }
}

// Shared piece: GEMM2 from staged h1 (K padded to 20, bias at k==18).
static __device__ __forceinline__ void gemm2(const float* __restrict__ stage,
                                             const v2f bf2[5], int wave, int lm,
                                             int half, v8f acc2[2]) {
#pragma unroll
  for (int t = 0; t < 2; t++) {
    int r = wave * 32 + t * 16 + lm;
#pragma unroll
    for (int kc = 0; kc < 5; kc++) {
      v2f a;
      a.x = stage[r * 21 + kc * 4 + half * 2 + 0];
      a.y = stage[r * 21 + kc * 4 + half * 2 + 1];
      acc2[t] = wmma4(a, bf2[kc], acc2[t]);
    }
  }
}

// ---------------------------------------------------------------------------
// Pass 2: GEMM1 + BN1 + relu -> GEMM2 -> z2 stats (9 channels)
// ---------------------------------------------------------------------------
__global__ void __launch_bounds__(256) k_pass2(
    const float* __restrict__ x, const float* __restrict__ emb,
    const float* __restrict__ W1, const float* __restrict__ b1,
    const float* __restrict__ W2, const float* __restrict__ b2,
    const float* __restrict__ S1, float* __restrict__ P2, long long B,
    int tiles) {
  __shared__ float stage[256 * 21];
  __shared__ float red[8 * 18];
  int tid = threadIdx.x, lane = tid & 31, wave = tid >> 5;
  int lm = lane & 15, half = lane >> 4;
  v2f bf1[3][2];
  buildB1(W1, b1, lm, half, bf1);
  v2f bf2[5];
  buildB2(W2, b2, lm, half, bf2);
  float sc1a = S1[lm], sh1a = S1[18 + lm];
  int n1 = 16 + lm;
  float sc1b = (n1 < 18) ? S1[n1] : 0.f;
  float sh1b = (n1 < 18) ? S1[18 + n1] : 0.f;
  float psum = 0.f, psq = 0.f;
  for (int tile = blockIdx.x; tile < tiles; tile += gridDim.x) {
    long long sbase0 = (long long)tile * 256;
    v8f acc[2][2] = {{{0,0,0,0,0,0,0,0},{0,0,0,0,0,0,0,0}},
                     {{0,0,0,0,0,0,0,0},{0,0,0,0,0,0,0,0}}};
    gemm1(x, emb, B, sbase0 + wave * 32, lm, half, bf1, acc);
    stageH1(stage, acc, sc1a, sh1a, sc1b, sh1b, sbase0, B, wave, lm, half);
    __syncthreads();
    v8f acc2[2] = {{0,0,0,0,0,0,0,0},{0,0,0,0,0,0,0,0}};
    gemm2(stage, bf2, wave, lm, half, acc2);
#pragma unroll
    for (int t = 0; t < 2; t++)
#pragma unroll
      for (int v = 0; v < 8; v++) {
        float z = acc2[t][v];  // zero for channels >= 9 and invalid rows
        psum += z;
        psq += z * z;
      }
    __syncthreads();  // stage reused next iteration
  }
  float s = psum + __shfl_xor(psum, 16, 32);
  float q = psq + __shfl_xor(psq, 16, 32);
  if (half == 0 && lm < 9) {
    red[wave * 18 + lm] = s;
    red[wave * 18 + 9 + lm] = q;
  }
  __syncthreads();
  if (tid < 18) {
    float tot = 0.f;
    for (int w = 0; w < 8; w++) tot += red[w * 18 + tid];
    P2[(long long)blockIdx.x * 18 + tid] = tot;
  }
}

// ---------------------------------------------------------------------------
// Pass 3: full pipeline + per-thread BiLSTM x2 + output
// ---------------------------------------------------------------------------
#define L1STEP(w, u, bb, xt)                                   \
  do {                                                         \
    float zi0 = w[0] * xt + u[0] * h0 + u[1] * h1 + bb[0];     \
    float zi1 = w[1] * xt + u[2] * h0 + u[3] * h1 + bb[1];     \
    float zf0 = w[2] * xt + u[4] * h0 + u[5] * h1 + bb[2];     \
    float zf1 = w[3] * xt + u[6] * h0 + u[7] * h1 + bb[3];     \
    float zg0 = w[4] * xt + u[8] * h0 + u[9] * h1 + bb[4];     \
    float zg1 = w[5] * xt + u[10] * h0 + u[11] * h1 + bb[5];   \
    float zo0 = w[6] * xt + u[12] * h0 + u[13] * h1 + bb[6];   \
    float zo1 = w[7] * xt + u[14] * h0 + u[15] * h1 + bb[7];   \
    c0 = sigf(zf0) * c0 + sigf(zi0) * tanhfast(zg0);           \
    c1 = sigf(zf1) * c1 + sigf(zi1) * tanhfast(zg1);           \
    h0 = sigf(zo0) * tanhfast(c0);                             \
    h1 = sigf(zo1) * tanhfast(c1);                             \
  } while (0)

#define L2STEP(w, u, bb, x0, x1, x2, x3)                                    \
  do {                                                                      \
    float zi = w[0] * x0 + w[1] * x1 + w[2] * x2 + w[3] * x3 + u[0] * hh + bb[0]; \
    float zf = w[4] * x0 + w[5] * x1 + w[6] * x2 + w[7] * x3 + u[1] * hh + bb[1]; \
    float zg = w[8] * x0 + w[9] * x1 + w[10] * x2 + w[11] * x3 + u[2] * hh + bb[2]; \
    float zo = w[12] * x0 + w[13] * x1 + w[14] * x2 + w[15] * x3 + u[3] * hh + bb[3]; \
    cc = sigf(zf) * cc + sigf(zi) * tanhfast(zg);                           \
    hh = sigf(zo) * tanhfast(cc);                                           \
  } while (0)

__global__ void __launch_bounds__(256) k_pass3(
    const float* __restrict__ x, const float* __restrict__ emb,
    const float* __restrict__ W1, const float* __restrict__ b1,
    const float* __restrict__ W2, const float* __restrict__ b2,
    const float* __restrict__ S1, const float* __restrict__ S2,
    const float* __restrict__ l1Wf, const float* __restrict__ l1Uf,
    const float* __restrict__ l1bf, const float* __restrict__ l1Wr,
    const float* __restrict__ l1Ur, const float* __restrict__ l1br,
    const float* __restrict__ l2Wf, const float* __restrict__ l2Uf,
    const float* __restrict__ l2bf, const float* __restrict__ l2Wr,
    const float* __restrict__ l2Ur, const float* __restrict__ l2br,
    float* __restrict__ out, long long B, int tiles) {
  __shared__ float stage[256 * 21];
  __shared__ float lw[112];
  int tid = threadIdx.x, lane = tid & 31, wave = tid >> 5;
  int lm = lane & 15, half = lane >> 4;
  if (tid < 112) {
    float v;
    if (tid < 8)        v = l1Wf[tid];
    else if (tid < 24)  v = l1Uf[tid - 8];
    else if (tid < 32)  v = l1bf[tid - 24];
    else if (tid < 40)  v = l1Wr[tid - 32];
    else if (tid < 56)  v = l1Ur[tid - 40];
    else if (tid < 64)  v = l1br[tid - 56];
    else if (tid < 80)  v = l2Wf[tid - 64];
    else if (tid < 84)  v = l2Uf[tid - 80];
    else if (tid < 88)  v = l2bf[tid - 84];
    else if (tid < 104) v = l2Wr[tid - 88];
    else if (tid < 108) v = l2Ur[tid - 104];
    else                v = l2br[tid - 108];
    lw[tid] = v;
  }
  v2f bf1[3][2];
  buildB1(W1, b1, lm, half, bf1);
  v2f bf2[5];
  buildB2(W2, b2, lm, half, bf2);
  float sc1a = S1[lm], sh1a = S1[18 + lm];
  int n1 = 16 + lm;
  float sc1b = (n1 < 18) ? S1[n1] : 0.f;
  float sh1b = (n1 < 18) ? S1[18 + n1] : 0.f;
  float sc2 = (lm < 9) ? S2[lm] : 0.f;
  float sh2 = (lm < 9) ? S2[9 + lm] : 0.f;
  __syncthreads();

  for (int tile = blockIdx.x; tile < tiles; tile += gridDim.x) {
    long long sbase0 = (long long)tile * 256;
    v8f acc[2][2] = {{{0,0,0,0,0,0,0,0},{0,0,0,0,0,0,0,0}},
                     {{0,0,0,0,0,0,0,0},{0,0,0,0,0,0,0,0}}};
    gemm1(x, emb, B, sbase0 + wave * 32, lm, half, bf1, acc);
    stageH1(stage, acc, sc1a, sh1a, sc1b, sh1b, sbase0, B, wave, lm, half);
    __syncthreads();
    v8f acc2[2] = {{0,0,0,0,0,0,0,0},{0,0,0,0,0,0,0,0}};
    gemm2(stage, bf2, wave, lm, half, acc2);
    __syncthreads();
    // BN2 + relu -> stage sequence (slots 0..8)
#pragma unroll
    for (int t = 0; t < 2; t++)
#pragma unroll
      for (int v = 0; v < 8; v++) {
        int r = wave * 32 + t * 16 + v + half * 8;
        if (lm < 9) {
          float val = acc2[t][v] * sc2 + sh2;
          stage[r * 21 + lm] = fmaxf(val, 0.f);
        }
      }
    __syncthreads();

    // ---- per-thread BiLSTM, sample s = sbase0 + tid ----
    long long s = sbase0 + tid;
    float seq[9];
#pragma unroll
    for (int k = 0; k < 9; k++) seq[k] = stage[tid * 21 + k];

    float y[9][4];
    {  // layer 1 forward
      const float* w = &lw[0];
      const float* u = &lw[8];
      const float* bb = &lw[24];
      float h0 = 0.f, h1 = 0.f, c0 = 0.f, c1 = 0.f;
#pragma unroll
      for (int t9 = 0; t9 < 9; t9++) {
        float xt = seq[t9];
        L1STEP(w, u, bb, xt);
        y[t9][0] = h0;
        y[t9][1] = h1;
      }
    }
    {  // layer 1 reverse
      const float* w = &lw[32];
      const float* u = &lw[40];
      const float* bb = &lw[56];
      float h0 = 0.f, h1 = 0.f, c0 = 0.f, c1 = 0.f;
#pragma unroll
      for (int t9 = 0; t9 < 9; t9++) {
        float xt = seq[8 - t9];
        L1STEP(w, u, bb, xt);
        y[8 - t9][2] = h0;
        y[8 - t9][3] = h1;
      }
    }
#pragma unroll
    for (int t9 = 0; t9 < 9; t9++)
#pragma unroll
      for (int j = 0; j < 4; j++) y[t9][j] = fmaxf(y[t9][j], 0.f);

    float outF[9], outR[9];
    {  // layer 2 forward
      const float* w = &lw[64];
      const float* u = &lw[80];
      const float* bb = &lw[84];
      float hh = 0.f, cc = 0.f;
#pragma unroll
      for (int t9 = 0; t9 < 9; t9++) {
        L2STEP(w, u, bb, y[t9][0], y[t9][1], y[t9][2], y[t9][3]);
        outF[t9] = hh;
      }
    }
    {  // layer 2 reverse
      const float* w = &lw[88];
      const float* u = &lw[104];
      const float* bb = &lw[108];
      float hh = 0.f, cc = 0.f;
#pragma unroll
      for (int t9 = 0; t9 < 9; t9++) {
        L2STEP(w, u, bb, y[8 - t9][0], y[8 - t9][1], y[8 - t9][2], y[8 - t9][3]);
        outR[8 - t9] = hh;
      }
    }
    if (s < B) {
      float* o = out + s * 18;
#pragma unroll
      for (int t9 = 0; t9 < 9; t9++) {
        o[t9 * 2 + 0] = outF[t9];
        o[t9 * 2 + 1] = outR[t9];
      }
    }
    __syncthreads();  // stage reused next iteration
  }
}

// ---------------------------------------------------------------------------
extern "C" void kernel_launch(void* const* d_in, const int* in_sizes, int n_in,
                              void* d_out, int out_size, void* d_ws,
                              size_t ws_size, hipStream_t stream) {
  (void)n_in;
  (void)out_size;
  const float* x    = (const float*)d_in[0];
  const float* emb  = (const float*)d_in[1];
  const float* W1   = (const float*)d_in[2];
  const float* b1   = (const float*)d_in[3];
  const float* g1   = (const float*)d_in[4];
  const float* be1  = (const float*)d_in[5];
  const float* W2   = (const float*)d_in[6];
  const float* b2   = (const float*)d_in[7];
  const float* g2   = (const float*)d_in[8];
  const float* be2  = (const float*)d_in[9];
  const float* l1Wf = (const float*)d_in[10];
  const float* l1Uf = (const float*)d_in[11];
  const float* l1bf = (const float*)d_in[12];
  const float* l1Wr = (const float*)d_in[13];
  const float* l1Ur = (const float*)d_in[14];
  const float* l1br = (const float*)d_in[15];
  const float* l2Wf = (const float*)d_in[16];
  const float* l2Uf = (const float*)d_in[17];
  const float* l2bf = (const float*)d_in[18];
  const float* l2Wr = (const float*)d_in[19];
  const float* l2Ur = (const float*)d_in[20];
  const float* l2br = (const float*)d_in[21];
  float* out = (float*)d_out;
  float* ws = (float*)d_ws;

  long long B = (long long)in_sizes[0] / 8;
  int tiles = (int)((B + 255) / 256);
  int nblk = 1024;
  if (nblk > tiles) nblk = tiles;
  long long wscap = (long long)(ws_size / 4);
  while (nblk > 1 && (64 + (long long)nblk * 54) > wscap) nblk /= 2;

  float* S1 = ws;        // 36 floats: scale1[18], shift1[18]
  float* S2 = ws + 36;   // 18 floats: scale2[9],  shift2[9]
  float* P1 = ws + 64;                         // nblk * 36
  float* P2 = ws + 64 + (long long)nblk * 36;  // nblk * 18

  k_pass1<<<nblk, 256, 0, stream>>>(x, emb, W1, b1, P1, B, tiles);
  k_fin<<<1, 32, 0, stream>>>(P1, g1, be1, S1, nblk, B, 18);
  k_pass2<<<nblk, 256, 0, stream>>>(x, emb, W1, b1, W2, b2, S1, P2, B, tiles);
  k_fin<<<1, 32, 0, stream>>>(P2, g2, be2, S2, nblk, B, 9);

  int nblk3 = tiles;
  if (nblk3 > 16384) nblk3 = 16384;
  k_pass3<<<nblk3, 256, 0, stream>>>(x, emb, W1, b1, W2, b2, S1, S2, l1Wf, l1Uf,
                                     l1bf, l1Wr, l1Ur, l1br, l2Wf, l2Uf, l2bf,
                                     l2Wr, l2Ur, l2br, out, B, tiles);
}